// ProductQuantizer_48284022342122
// MI455X (gfx1250) — compile-verified
//
#include <hip/hip_runtime.h>

typedef __attribute__((ext_vector_type(2))) float v2f;
typedef __attribute__((ext_vector_type(8))) float v8f;

#define BETA_F    0.25f
#define B_TOT     4096
#define M_TOT     64
#define D_DIM     64
#define K_CODES   256
#define TILE_B    128
#define LDS_STRIDE 68   // 64 + 4 pad: 272B rows -> 16B aligned, conflict-free frag reads

// LDS partition (floats):
//   lds_a : TILE_B  * LDS_STRIDE   (z tile, 128 rows x 64 d)
//   lds_c : K_CODES * LDS_STRIDE   (codebook m, 256 rows x 64 d)
//   c2_s  : K_CODES                (per-code squared norms)
//   idx_s : TILE_B                 (winning code per row, int bits)
//   red   : 256                    (block reduction scratch)
#define SMEM_FLOATS (TILE_B*LDS_STRIDE + K_CODES*LDS_STRIDE + K_CODES + TILE_B + 256)

__global__ __launch_bounds__(256) void pq_main_kernel(
    const float* __restrict__ z,        // (4096, 64, 64)
    const float* __restrict__ cb,       // (64, 256, 64)
    float* __restrict__ out_q,          // (4096, 64, 64)
    float* __restrict__ out_idx,        // (4096, 64) as float
    float* __restrict__ partials)       // (gridDim.x) partial loss sums
{
    extern __shared__ float smem[];
    float* lds_a = smem;
    float* lds_c = lds_a + TILE_B * LDS_STRIDE;
    float* c2_s  = lds_c + K_CODES * LDS_STRIDE;
    int*   idx_s = (int*)(c2_s + K_CODES);
    float* red   = (float*)(idx_s + TILE_B);

    const int m     = blockIdx.x & (M_TOT - 1);
    const int btile = blockIdx.x >> 6;
    const int bbase = btile * TILE_B;
    const int tid   = threadIdx.x;       // 0..255
    const int lane  = tid & 31;
    const int wave  = tid >> 5;          // 0..7

    // ---------------- stage z tile and codebook slice into LDS ----------------
    // z tile: 128 rows x 64 floats = 2048 float4, 8 per thread
    #pragma unroll
    for (int it = 0; it < 8; ++it) {
        int g  = it * 256 + tid;
        int r  = g >> 4;
        int c4 = g & 15;
        float4 v = ((const float4*)(z + ((size_t)(bbase + r) * M_TOT + m) * D_DIM))[c4];
        *(float4*)(lds_a + r * LDS_STRIDE + c4 * 4) = v;
    }
    // codebook m: 256 rows x 64 floats = 4096 float4, 16 per thread
    #pragma unroll
    for (int it = 0; it < 16; ++it) {
        int g  = it * 256 + tid;
        int k  = g >> 4;
        int c4 = g & 15;
        float4 v = ((const float4*)(cb + ((size_t)m * K_CODES + k) * D_DIM))[c4];
        *(float4*)(lds_c + k * LDS_STRIDE + c4 * 4) = v;
    }
    __syncthreads();

    // ---------------- c2[k] = ||codebook row k||^2 ----------------
    {
        const float* row = lds_c + tid * LDS_STRIDE;
        float s = 0.f;
        #pragma unroll
        for (int d = 0; d < D_DIM; ++d) s = fmaf(row[d], row[d], s);
        c2_s[tid] = s;
    }
    __syncthreads();

    // ---------------- WMMA: zc = Z_tile x Cb^T, fused argmin(c2 - 2*zc) ------
    const int half = lane >> 4;    // 0: lanes 0-15, 1: lanes 16-31
    const int l16  = lane & 15;

    // A fragment (16x4 f32 per step): lanes 0-15 hold K={0,1}, 16-31 K={2,3}
    const int rowA = wave * 16 + l16;
    v2f afrag[16];
    #pragma unroll
    for (int ks = 0; ks < 16; ++ks)
        afrag[ks] = *(const v2f*)(lds_a + rowA * LDS_STRIDE + ks * 4 + 2 * half);

    float minval[8];
    int   minidx[8];
    #pragma unroll
    for (int i = 0; i < 8; ++i) { minval[i] = __builtin_inff(); minidx[i] = 0; }

    for (int kt = 0; kt < 16; ++kt) {        // 16-wide tiles over K=256 codes
        v8f acc = {};
        #pragma unroll
        for (int ks = 0; ks < 16; ++ks) {    // d = 64 in chunks of 4
            // B fragment (4x16 f32): B[kd][n] = cb[code n][kd]
            v2f bfrag = *(const v2f*)(lds_c + (kt * 16 + l16) * LDS_STRIDE
                                      + ks * 4 + 2 * half);
            acc = __builtin_amdgcn_wmma_f32_16x16x4_f32(
                /*neg_a=*/false, afrag[ks], /*neg_b=*/false, bfrag,
                /*c_mod=*/(short)0, acc, /*reuse_a=*/false, /*reuse_b=*/false);
        }
        // score = c2[col] - 2*zc ; argmin over k is independent of z2
        const int   colg = kt * 16 + l16;
        const float c2v  = c2_s[colg];
        #pragma unroll
        for (int i = 0; i < 8; ++i) {        // acc[i] -> row (wave*16 + i + 8*half)
            float score = fmaf(-2.f, acc[i], c2v);
            if (score < minval[i]) { minval[i] = score; minidx[i] = colg; }
        }
    }

    // ---------------- cross-lane argmin over the 16 lanes of each half -------
    #pragma unroll
    for (int i = 0; i < 8; ++i) {
        float v  = minval[i];
        int   ix = minidx[i];
        #pragma unroll
        for (int off = 8; off >= 1; off >>= 1) {   // masks <=8 never cross halves
            float ov = __shfl_xor(v,  off, 32);
            int   oi = __shfl_xor(ix, off, 32);
            if (ov < v || (ov == v && oi < ix)) { v = ov; ix = oi; }
        }
        if (l16 == 0) {
            int rowT = wave * 16 + i + 8 * half;
            idx_s[rowT] = ix;
            out_idx[(size_t)(bbase + rowT) * M_TOT + m] = (float)ix;
        }
    }
    __syncthreads();

    // ---------------- gather winning rows + squared-error accumulation ------
    float lsum = 0.f;
    #pragma unroll
    for (int it = 0; it < 8; ++it) {
        int g  = it * 256 + tid;
        int r  = g >> 4;
        int c4 = g & 15;
        int k  = idx_s[r];
        float4 q  = *(const float4*)(lds_c + k * LDS_STRIDE + c4 * 4);
        float4 zv = *(const float4*)(lds_a + r * LDS_STRIDE + c4 * 4);
        *(float4*)(out_q + ((size_t)(bbase + r) * M_TOT + m) * D_DIM + c4 * 4) = q;
        float dx = q.x - zv.x, dy = q.y - zv.y, dz = q.z - zv.z, dw = q.w - zv.w;
        lsum += dx * dx + dy * dy + dz * dz + dw * dw;
    }

    // deterministic block reduction (fixed order, no atomics)
    red[tid] = lsum;
    __syncthreads();
    #pragma unroll
    for (int s = 128; s > 0; s >>= 1) {
        if (tid < s) red[tid] += red[tid + s];
        __syncthreads();
    }
    if (tid == 0) partials[blockIdx.x] = red[0];
}

__global__ __launch_bounds__(256) void pq_reduce_kernel(
    const float* __restrict__ partials, float* __restrict__ out_loss)
{
    __shared__ float red[256];
    const int tid = threadIdx.x;
    float s = 0.f;
    #pragma unroll
    for (int j = 0; j < 8; ++j) s += partials[j * 256 + tid];  // fixed order
    red[tid] = s;
    __syncthreads();
    #pragma unroll
    for (int st = 128; st > 0; st >>= 1) {
        if (tid < st) red[tid] += red[tid + st];
        __syncthreads();
    }
    if (tid == 0) {
        // q_loss = (1 + BETA) * mean((q - z)^2), mean over 4096*64*64 elements
        *out_loss = (1.0f + BETA_F) * red[0] * (1.0f / 16777216.0f);
    }
}

extern "C" void kernel_launch(void* const* d_in, const int* in_sizes, int n_in,
                              void* d_out, int out_size, void* d_ws, size_t ws_size,
                              hipStream_t stream)
{
    (void)in_sizes; (void)n_in; (void)out_size; (void)ws_size;
    const float* z  = (const float*)d_in[0];   // (4096, 64, 64)
    const float* cb = (const float*)d_in[1];   // (64, 256, 64)

    float* out      = (float*)d_out;
    float* out_q    = out;                                      // 16,777,216
    float* out_idx  = out_q   + (size_t)B_TOT * M_TOT * D_DIM;  //    262,144
    float* out_loss = out_idx + (size_t)B_TOT * M_TOT;          //          1
    float* partials = (float*)d_ws;                             // 2048 floats

    const size_t smem_bytes = (size_t)SMEM_FLOATS * sizeof(float); // ~105 KB (<320 KB WGP LDS)
    hipFuncSetAttribute((const void*)pq_main_kernel,
                        hipFuncAttributeMaxDynamicSharedMemorySize, (int)smem_bytes);

    const int nblocks = (B_TOT / TILE_B) * M_TOT;  // 32 * 64 = 2048
    pq_main_kernel<<<nblocks, 256, smem_bytes, stream>>>(z, cb, out_q, out_idx, partials);
    pq_reduce_kernel<<<1, 256, 0, stream>>>(partials, out_loss);
}